// LayerSumSimple_43164421325297
// MI455X (gfx1250) — compile-verified
//
#include <hip/hip_runtime.h>
#include <stdint.h>

// Problem constants (match reference): B=8, T=8192, D=256, K=8.
constexpr int Bb   = 8;
constexpr int T    = 8192;
constexpr int D    = 256;      // channels == threads per block (8 wave32 waves)
constexpr int KK   = 8;
constexpr int NT   = T - (KK - 1);   // 8185 output timesteps
constexpr int C    = 64;             // time chunks (parallelism: 2048*64 lanes)
constexpr int L    = T / C;          // 128 arrivals per chunk
constexpr int TILE = 32;             // arrivals per LDS tile (32*256*4 = 32 KiB)
constexpr int NTILES = L / TILE;     // 4

// Scratch layout (floats):
//   ops   : [C][Bb][36][D]   (28 strictly-lower A entries + 8 v entries) ~18 MiB
//   states: [C][Bb][8][D]                                                ~4 MiB
constexpr size_t OPS_FLOATS = (size_t)C * Bb * 36 * D;

// ---------------------------------------------------------------------------
// CDNA5 async copy: global -> LDS, 128b per lane, GVS addressing.
// One call moves one TILE (TILE*D floats = 32 KiB): 8 b128 quads per lane,
// consecutive lanes touch consecutive 16B quads (fully coalesced).
// ---------------------------------------------------------------------------
__device__ __forceinline__ void async_load_tile(const float* gsrc, float* lds_dst) {
  unsigned lds_base = (unsigned)(uintptr_t)lds_dst;   // low 32 bits = LDS byte addr
  unsigned long long gbase = (unsigned long long)(uintptr_t)gsrc;
#pragma unroll
  for (int j = 0; j < 8; ++j) {
    unsigned byteoff  = (unsigned)((j * D + threadIdx.x) * 16u);
    unsigned lds_addr = lds_base + byteoff;
    asm volatile("global_load_async_to_lds_b128 %0, %1, %2"
                 :: "v"(lds_addr), "v"(byteoff), "s"(gbase)
                 : "memory");
  }
}

__device__ __forceinline__ void wait_async0() {
  asm volatile("s_wait_asynccnt 0" ::: "memory");
}

// ---------------------------------------------------------------------------
// K1: per (chunk c, batch b, channel d) compose the chunk's max-plus operator.
// Per arrival u:  y_k = fma(x_u, w_k, b_k)
//   v[k]    = max(v[k],    y_k + v[k-1])        (k = 7..1, old values)
//   A[k][j] = max(A[k][j], y_k + A[k-1][j])     (j < k)
//   v[0]    = max(v[0],    y_0)
// A starts as the max-plus identity; its diagonal stays exactly 0 forever, so
// only the 28 strictly-lower entries + 8 v entries are live/stored.
// ---------------------------------------------------------------------------
__global__ __launch_bounds__(256) void k1_ops(const float* __restrict__ X,
                                              const float* __restrict__ W,
                                              const float* __restrict__ Bp,
                                              float* __restrict__ ops) {
  const int c = blockIdx.x, b = blockIdx.y, d = threadIdx.x;
  __shared__ __align__(16) float tile[2][TILE * D];

  float wk[KK], bk[KK];
#pragma unroll
  for (int k = 0; k < KK; ++k) { wk[k] = W[k * D + d]; bk[k] = Bp[k * D + d]; }

  const float NI = -__builtin_inff();
  float A[KK][KK], v[KK];
#pragma unroll
  for (int k = 0; k < KK; ++k) {
    v[k] = NI;
#pragma unroll
    for (int j = 0; j < KK; ++j) A[k][j] = (k == j) ? 0.0f : NI;
  }

  const float* xbase = X + ((size_t)b * T + (size_t)c * L) * D;

  async_load_tile(xbase, &tile[0][0]);
  wait_async0();
  __syncthreads();

  for (int ti = 0; ti < NTILES; ++ti) {
    const int cur = ti & 1;
    if (ti + 1 < NTILES)
      async_load_tile(xbase + (size_t)(ti + 1) * TILE * D, &tile[cur ^ 1][0]);
#pragma unroll 8
    for (int u = 0; u < TILE; ++u) {
      const float x = tile[cur][u * D + d];
      float y[KK];
#pragma unroll
      for (int k = 0; k < KK; ++k) y[k] = fmaf(x, wk[k], bk[k]);
#pragma unroll
      for (int k = KK - 1; k >= 1; --k) {
        v[k] = fmaxf(v[k], y[k] + v[k - 1]);
#pragma unroll
        for (int j = 0; j < k; ++j)
          A[k][j] = fmaxf(A[k][j], y[k] + A[k - 1][j]);
      }
      v[0] = fmaxf(v[0], y[0]);
    }
    wait_async0();
    __syncthreads();
  }

  float* ob = ops + (size_t)(c * Bb + b) * 36 * D + d;
#pragma unroll
  for (int k = 1; k < KK; ++k)
#pragma unroll
    for (int j = 0; j < k; ++j)
      ob[(size_t)(k * (k - 1) / 2 + j) * D] = A[k][j];
#pragma unroll
  for (int k = 0; k < KK; ++k) ob[(size_t)(28 + k) * D] = v[k];
}

// ---------------------------------------------------------------------------
// K2 helpers: 36-entry operator load / one chunk composition step.
// ---------------------------------------------------------------------------
__device__ __forceinline__ void k2_load(float (&a)[36], const float* __restrict__ ops,
                                        int c, int b, int d) {
#pragma unroll
  for (int e = 0; e < 36; ++e)
    a[e] = ops[((size_t)(c * Bb + b) * 36 + e) * D + d];
}

__device__ __forceinline__ void k2_step(float (&S)[KK], const float (&a)[36],
                                        float* __restrict__ states,
                                        int c, int b, int d) {
#pragma unroll
  for (int k = 0; k < KK; ++k)                  // record entry state of chunk c
    states[((size_t)(c * Bb + b) * KK + k) * D + d] = S[k];
  float Sn[KK];
#pragma unroll
  for (int k = 0; k < KK; ++k) {
    float t = fmaxf(S[k], a[28 + k]);           // diagonal (==0) + v_k
#pragma unroll
    for (int j = 0; j < k; ++j)
      t = fmaxf(t, a[k * (k - 1) / 2 + j] + S[j]);
    Sn[k] = t;
  }
#pragma unroll
  for (int k = 0; k < KK; ++k) S[k] = Sn[k];
}

// ---------------------------------------------------------------------------
// K2: serial over chunks (parallel over the 2048 (b,d) lanes): propagate the
// entry state through each chunk operator. 4 rotating register buffers give
// a prefetch distance of ~3-4 chunks to hide (L2-resident) load latency on
// this latency-critical middle phase.
// ---------------------------------------------------------------------------
__global__ __launch_bounds__(256) void k2_compose(const float* __restrict__ ops,
                                                  float* __restrict__ states) {
  const int b = blockIdx.x, d = threadIdx.x;
  const float NI = -__builtin_inff();
  float S[KK];
#pragma unroll
  for (int k = 0; k < KK; ++k) S[k] = NI;

  float B0[36], B1[36], B2[36], B3[36];
  k2_load(B0, ops, 0, b, d);
  k2_load(B1, ops, 1, b, d);
  k2_load(B2, ops, 2, b, d);
  k2_load(B3, ops, 3, b, d);

  for (int c = 0; c < C; c += 4) {              // C % 4 == 0
    k2_step(S, B0, states, c + 0, b, d);
    if (c + 4 < C) k2_load(B0, ops, c + 4, b, d);
    k2_step(S, B1, states, c + 1, b, d);
    if (c + 5 < C) k2_load(B1, ops, c + 5, b, d);
    k2_step(S, B2, states, c + 2, b, d);
    if (c + 6 < C) k2_load(B2, ops, c + 6, b, d);
    k2_step(S, B3, states, c + 3, b, d);
    if (c + 7 < C) k2_load(B3, ops, c + 7, b, d);
  }
}

// ---------------------------------------------------------------------------
// K3: parallel over chunks again: run the exact per-arrival scan from the
// chunk's true entry state and emit outputs. X streamed via async LDS tiles.
// The u>=7 output guard is hoisted per tile: only chunk 0 / tile 0 needs it.
// ---------------------------------------------------------------------------
__global__ __launch_bounds__(256) void k3_scan(const float* __restrict__ X,
                                               const float* __restrict__ W,
                                               const float* __restrict__ Bp,
                                               const float* __restrict__ states,
                                               float* __restrict__ out) {
  const int c = blockIdx.x, b = blockIdx.y, d = threadIdx.x;
  __shared__ __align__(16) float tile[2][TILE * D];

  float wk[KK], bk[KK], S[KK];
#pragma unroll
  for (int k = 0; k < KK; ++k) {
    wk[k] = W[k * D + d];
    bk[k] = Bp[k * D + d];
    S[k]  = states[((size_t)(c * Bb + b) * KK + k) * D + d];
  }

  const float* xbase = X + ((size_t)b * T + (size_t)c * L) * D;
  float* ob = out + (size_t)b * NT * D + d;

  async_load_tile(xbase, &tile[0][0]);
  wait_async0();
  __syncthreads();

  for (int ti = 0; ti < NTILES; ++ti) {
    const int cur = ti & 1;
    if (ti + 1 < NTILES)
      async_load_tile(xbase + (size_t)(ti + 1) * TILE * D, &tile[cur ^ 1][0]);

    const int ubase = c * L + ti * TILE;             // uniform
    if (ubase >= KK - 1) {
      // steady state: every arrival emits an output (branchless stores)
#pragma unroll 8
      for (int u = 0; u < TILE; ++u) {
        const float x = tile[cur][u * D + d];
        float y[KK];
#pragma unroll
        for (int k = 0; k < KK; ++k) y[k] = fmaf(x, wk[k], bk[k]);
#pragma unroll
        for (int k = KK - 1; k >= 1; --k)
          S[k] = fmaxf(S[k], S[k - 1] + y[k]);
        S[0] = fmaxf(S[0], y[0]);
        ob[(size_t)(ubase + u - (KK - 1)) * D] = S[KK - 1];
      }
    } else {
      // warm-up tile (chunk 0, tile 0 only): guard the first 7 arrivals
#pragma unroll 8
      for (int u = 0; u < TILE; ++u) {
        const int ug = ubase + u;
        const float x = tile[cur][u * D + d];
        float y[KK];
#pragma unroll
        for (int k = 0; k < KK; ++k) y[k] = fmaf(x, wk[k], bk[k]);
#pragma unroll
        for (int k = KK - 1; k >= 1; --k)
          S[k] = fmaxf(S[k], S[k - 1] + y[k]);
        S[0] = fmaxf(S[0], y[0]);
        if (ug >= KK - 1)
          ob[(size_t)(ug - (KK - 1)) * D] = S[KK - 1];
      }
    }
    wait_async0();
    __syncthreads();
  }
}

// ---------------------------------------------------------------------------
extern "C" void kernel_launch(void* const* d_in, const int* in_sizes, int n_in,
                              void* d_out, int out_size, void* d_ws, size_t ws_size,
                              hipStream_t stream) {
  (void)in_sizes; (void)n_in; (void)out_size; (void)ws_size;
  const float* X  = (const float*)d_in[0];   // (B,T,D) f32
  const float* W  = (const float*)d_in[1];   // (K,D)   f32
  const float* Bp = (const float*)d_in[2];   // (K,D)   f32
  float* out = (float*)d_out;                // (B,T-K+1,D) f32

  float* ops    = (float*)d_ws;
  float* states = ops + OPS_FLOATS;          // ws needs ~22 MiB total

  k1_ops    <<<dim3(C, Bb), D, 0, stream>>>(X, W, Bp, ops);
  k2_compose<<<dim3(Bb),    D, 0, stream>>>(ops, states);
  k3_scan   <<<dim3(C, Bb), D, 0, stream>>>(X, W, Bp, states, out);
}